// YOLO12_postprocess_5514738008445
// MI455X (gfx1250) — compile-verified
//
#include <hip/hip_runtime.h>

#define N_ANCHORS 8400
#define N_PAD     16384          // next pow2 for bitonic sort
#define NWORDS    263            // ceil(8400/32)
#define SCORE_TH  0.5f

typedef unsigned int  v4u __attribute__((ext_vector_type(4)));
typedef int           v4i __attribute__((ext_vector_type(4)));
typedef int           v8i __attribute__((ext_vector_type(8)));

// TDM builtins exist only in the device pass; host pass must parse the fallback.
#if defined(__has_builtin)
#  if __has_builtin(__builtin_amdgcn_tensor_load_to_lds) && __has_builtin(__builtin_amdgcn_s_wait_tensorcnt)
#    define HAVE_TDM 1
#  else
#    define HAVE_TDM 0
#  endif
#else
#  define HAVE_TDM 0
#endif

// Positive confirmation: device compilation MUST have the gfx1250 TDM path.
#if defined(__HIP_DEVICE_COMPILE__) && !HAVE_TDM
#  error "gfx1250 TDM builtins not available in device compilation"
#endif

// ---------------------------------------------------------------- workspace map
__host__ __device__ constexpr size_t al256(size_t x) { return (x + 255) & ~(size_t)255; }
constexpr size_t OFF_KEY   = 0;                                          // u64[16384]
constexpr size_t OFF_XYXY  = al256(OFF_KEY   + (size_t)N_PAD * 8);       // float4[8400]
constexpr size_t OFF_SCORE = al256(OFF_XYXY  + (size_t)N_ANCHORS * 16);  // f32[8400]
constexpr size_t OFF_VALID = al256(OFF_SCORE + (size_t)N_ANCHORS * 4);   // u32[8400]
constexpr size_t OFF_BOXS  = al256(OFF_VALID + (size_t)N_ANCHORS * 4);   // float4[8400] sorted
constexpr size_t OFF_SCRS  = al256(OFF_BOXS  + (size_t)N_ANCHORS * 16);  // f32[8400] sorted
constexpr size_t OFF_VALS  = al256(OFF_SCRS  + (size_t)N_ANCHORS * 4);   // u32[8400] sorted
constexpr size_t OFF_KEEPW = al256(OFF_VALS  + (size_t)N_ANCHORS * 4);   // u32[263]
constexpr size_t OFF_MASK  = al256(OFF_KEEPW + (size_t)NWORDS * 4);      // u32[8400*263]
constexpr size_t WS_FULL   = OFF_MASK + (size_t)N_ANCHORS * NWORDS * 4;

// ---------------------------------------------------------------- TDM helper
#if HAVE_TDM
// 1D contiguous f32 copy: global -> LDS via Tensor Data Mover.
// tensor_elems = elements remaining from gsrc (OOB reads return zero),
// tile_elems   = elements to transfer (fits 16-bit tile_dim0, multiple of 4B).
__device__ __forceinline__ void tdm_load_f32(unsigned lds_byte_off, const float* gsrc,
                                             unsigned tile_elems, unsigned tensor_elems) {
  unsigned long long ga = (unsigned long long)(const void*)gsrc;
  v4u g0;
  g0[0] = 1u;                                                  // count=1, user mode
  g0[1] = lds_byte_off;                                        // lds_addr
  g0[2] = (unsigned)ga;                                        // global_addr[31:0]
  g0[3] = ((unsigned)(ga >> 32) & 0x01FFFFFFu) | 0x80000000u;  // global_addr[56:32] | type=2
  v8i g1;
  g1[0] = (int)(2u << 16);                                     // wg_mask=0, data_size=4B
  g1[1] = (int)((tensor_elems & 0xFFFFu) << 16);               // tensor_dim0[15:0]
  g1[2] = (int)(((tensor_elems >> 16) & 0xFFFFu) | (1u << 16));// tensor_dim0[31:16], tensor_dim1=1
  g1[3] = (int)((tile_elems & 0xFFFFu) << 16);                 // tile_dim0
  g1[4] = 1;                                                   // tile_dim1=1, tile_dim2=0
  g1[5] = (int)tensor_elems;                                   // tensor_dim0_stride[31:0]
  g1[6] = 0;
  g1[7] = 0;
  v4i z4 = 0;
#if defined(__clang_major__) && (__clang_major__ >= 23)
  v8i z8 = 0;
  __builtin_amdgcn_tensor_load_to_lds(g0, g1, z4, z4, z8, 0);
#else
  __builtin_amdgcn_tensor_load_to_lds(g0, g1, z4, z4, 0);
#endif
}
#endif

__device__ __forceinline__ unsigned lds_off_of(const void* p) {
  // LDS aperture: flat addr[31:0] is the LDS byte offset.
  return (unsigned)(unsigned long long)p;
}

// ---------------------------------------------------------------- IoU predicate
__device__ __forceinline__ bool iou_gt_half(const float4 a, const float4 b) {
  float areaA = (a.z - a.x) * (a.w - a.y);
  float areaB = (b.z - b.x) * (b.w - b.y);
  float ltx = fmaxf(a.x, b.x), lty = fmaxf(a.y, b.y);
  float rbx = fminf(a.z, b.z), rby = fminf(a.w, b.w);
  float iw = fmaxf(rbx - ltx, 0.0f), ih = fmaxf(rby - lty, 0.0f);
  float inter = iw * ih;
  float uni = areaA + areaB - inter;
  return (inter / uni) > 0.5f;   // NaN compares false, matching jnp semantics
}

// ---------------------------------------------------------------- kernel 1: prep
// Transpose channels, compute xyxy, validity, and sortable 64-bit keys.
// Ascending sort of key == stable descending argsort of masked score.
__global__ void k_prep(const float* __restrict__ in, unsigned long long* __restrict__ key,
                       float4* __restrict__ xyxy, float* __restrict__ score,
                       unsigned* __restrict__ valid) {
  unsigned i = blockIdx.x * blockDim.x + threadIdx.x;
  if (i >= N_PAD) return;
  if (i < N_ANCHORS) {
    float cx = in[0 * N_ANCHORS + i];
    float cy = in[1 * N_ANCHORS + i];
    float w  = in[2 * N_ANCHORS + i];
    float h  = in[3 * N_ANCHORS + i];
    float s  = in[4 * N_ANCHORS + i];
    unsigned v = (s > SCORE_TH) ? 1u : 0u;
    float ms = v ? s : -__builtin_inff();
    unsigned u = __float_as_uint(ms);
    u = (u & 0x80000000u) ? ~u : (u | 0x80000000u);   // monotone float->uint
    key[i] = ((unsigned long long)(~u) << 32) | (unsigned long long)i;
    float hw = w * 0.5f, hh = h * 0.5f;
    xyxy[i]  = make_float4(cx - hw, cy - hh, cx + hw, cy + hh);
    score[i] = s;
    valid[i] = v;
  } else {
    key[i] = ~0ULL;                                    // pad sorts last
  }
}

// ---------------------------------------------------------------- kernel 2a: global bitonic pass (j > 256)
__global__ void k_bitonic(unsigned long long* __restrict__ key, unsigned j, unsigned k) {
  unsigned t = blockIdx.x * blockDim.x + threadIdx.x;   // 0..N_PAD-1
  unsigned ixj = t ^ j;
  if (ixj > t) {
    unsigned long long a = key[t], b = key[ixj];
    bool up = ((t & k) == 0);
    if ((a > b) == up) { key[t] = b; key[ixj] = a; }
  }
}

// ---------------------------------------------------------------- kernel 2b: fused LDS bitonic tail
// All passes with j <= 256 stay inside an aligned 512-element window: run them
// in LDS with one launch. 256 threads = 256 comparators per pass, 8 wave32s.
__global__ void k_bitonic_fused(unsigned long long* __restrict__ key, unsigned jstart, unsigned k) {
  __shared__ unsigned long long s[512];
  const unsigned base = blockIdx.x * 512u;
  const unsigned t = threadIdx.x;            // 0..255
  s[t]        = key[base + t];
  s[t + 256]  = key[base + t + 256];
  __syncthreads();
  for (unsigned j = jstart; j > 0; j >>= 1) {
    unsigned e = (t & (j - 1u)) + ((t & ~(j - 1u)) << 1);   // lower element of comparator
    unsigned long long a = s[e], b = s[e + j];
    bool up = (((base + e) & k) == 0);        // j < k -> direction uniform per comparator
    if ((a > b) == up) { s[e] = b; s[e + j] = a; }
    __syncthreads();
  }
  key[base + t]       = s[t];
  key[base + t + 256] = s[t + 256];
}

// ---------------------------------------------------------------- kernel 3: gather sorted
__global__ void k_gather(const unsigned long long* __restrict__ key,
                         const float4* __restrict__ xyxy, const float* __restrict__ score,
                         const unsigned* __restrict__ valid,
                         float4* __restrict__ boxs, float* __restrict__ scrs,
                         unsigned* __restrict__ vals) {
  unsigned i = blockIdx.x * blockDim.x + threadIdx.x;
  if (i >= N_ANCHORS) return;
  unsigned idx = (unsigned)(key[i] & 0xFFFFFFFFu);
  unsigned v = valid[idx];
  boxs[i] = xyxy[idx];
  vals[i] = v;
  scrs[i] = v ? score[idx] : 0.0f;   // where(valid, scores, 0)[order]
}

// ---------------------------------------------------------------- kernel 4: suppression bitmask
// Block: 64 i-rows x 256 j-cols. Tiles of sorted boxes staged to LDS by the TDM.
// Sub-diagonal tiles (all j <= i) are all-zero: write zeros, skip loads + IoU.
#define BI 64
#define BJ 256
__global__ void k_mask(const float4* __restrict__ boxs, unsigned* __restrict__ mask) {
  __shared__ float4 sI[BI];
  __shared__ float4 sJ[BJ];
  const unsigned i0 = blockIdx.x * BI;
  const unsigned j0 = blockIdx.y * BJ;

  const bool all_zero = (j0 + BJ - 1u) <= i0;   // max j in tile <= min i in tile
  if (!all_zero) {
#if HAVE_TDM
    if (threadIdx.x < 32) {           // one wave issues both TDM descriptors
      tdm_load_f32(lds_off_of(sI), (const float*)(boxs + i0), BI * 4, (N_ANCHORS - i0) * 4);
      tdm_load_f32(lds_off_of(sJ), (const float*)(boxs + j0), BJ * 4, (N_ANCHORS - j0) * 4);
      __builtin_amdgcn_s_wait_tensorcnt(0);
    }
#else
    for (unsigned t = threadIdx.x; t < BI; t += blockDim.x) {
      unsigned g = i0 + t; sI[t] = boxs[g < N_ANCHORS ? g : (N_ANCHORS - 1)];
    }
    for (unsigned t = threadIdx.x; t < BJ; t += blockDim.x) {
      unsigned g = j0 + t; sJ[t] = boxs[g < N_ANCHORS ? g : (N_ANCHORS - 1)];
    }
#endif
    __syncthreads();
  }

  #pragma unroll
  for (int p = 0; p < 2; ++p) {
    unsigned pidx = threadIdx.x + (unsigned)p * 256u;  // 0..511 -> (i_local, word_local)
    unsigned il = pidx >> 3;
    unsigned wl = pidx & 7u;
    unsigned i  = i0 + il;
    unsigned word = (j0 >> 5) + wl;
    if (i < N_ANCHORS && word < NWORDS) {
      unsigned bits = 0u;
      if (!all_zero) {
        float4 bi = sI[il];
        for (int b = 0; b < 32; ++b) {
          unsigned j = (word << 5) + (unsigned)b;
          if (j < N_ANCHORS && j > i) {
            if (iou_gt_half(bi, sJ[wl * 32u + (unsigned)b])) bits |= (1u << b);
          }
        }
      }
      mask[(size_t)i * NWORDS + word] = bits;
    }
  }
}

// ---------------------------------------------------------------- kernel 5: serial greedy scan
// Exactly ONE wave32: LDS ops within a wave are in-order, no barriers needed in loop.
__global__ void k_scan(const unsigned* __restrict__ mask, const unsigned* __restrict__ vals,
                       unsigned* __restrict__ keepw) {
  volatile __shared__ unsigned removed[NWORDS];
  __shared__ unsigned validw[NWORDS];
  const int lane = threadIdx.x;
  for (int w = lane; w < NWORDS; w += 32) {
    removed[w] = 0u;
    unsigned v = 0u;
    for (int b = 0; b < 32; ++b) {
      int j = w * 32 + b;
      if (j < N_ANCHORS && vals[j]) v |= (1u << b);
    }
    validw[w] = v;
  }
  __syncthreads();
  for (int i = 0; i < N_ANCHORS; ++i) {
    unsigned rw = removed[i >> 5];
    unsigned vw = validw[i >> 5];
    bool live = (((vw >> (i & 31)) & 1u) != 0u) && (((rw >> (i & 31)) & 1u) == 0u);
    if (live) {                                   // uniform across the wave
      const unsigned* row = mask + (size_t)i * NWORDS;
      for (int w = lane; w < NWORDS; w += 32) removed[w] |= row[w];
    }
  }
  for (int w = lane; w < NWORDS; w += 32) keepw[w] = validw[w] & ~removed[w];
}

// ---------------------------------------------------------------- fallback: direct NMS (small ws)
__global__ void k_nms_direct(const float4* __restrict__ boxs, const unsigned* __restrict__ vals,
                             unsigned* __restrict__ keepw) {
  __shared__ unsigned removed[NWORDS];
  __shared__ float4 cur;
  __shared__ int live;
  const int tid = threadIdx.x;
  for (int w = tid; w < NWORDS; w += (int)blockDim.x) removed[w] = 0u;
  __syncthreads();
  for (int i = 0; i < N_ANCHORS; ++i) {
    if (tid == 0) {
      bool l = vals[i] && !((removed[i >> 5] >> (i & 31)) & 1u);
      live = l ? 1 : 0;
      if (l) cur = boxs[i];
    }
    __syncthreads();
    if (live) {
      float4 bi = cur;
      for (int j = i + 1 + tid; j < N_ANCHORS; j += (int)blockDim.x) {
        if (iou_gt_half(bi, boxs[j])) atomicOr((unsigned*)&removed[j >> 5], 1u << (j & 31));
      }
    }
    __syncthreads();
  }
  for (int w = tid; w < NWORDS; w += (int)blockDim.x) {
    unsigned v = 0u;
    for (int b = 0; b < 32; ++b) {
      int j = w * 32 + b;
      if (j < N_ANCHORS && vals[j]) v |= (1u << b);
    }
    keepw[w] = v & ~removed[w];
  }
}

// ---------------------------------------------------------------- kernel 6: emit output
__global__ void k_write(const float4* __restrict__ boxs, const float* __restrict__ scrs,
                        const unsigned* __restrict__ keepw, float* __restrict__ out) {
  unsigned i = blockIdx.x * blockDim.x + threadIdx.x;
  if (i >= N_ANCHORS) return;
  bool kept = ((keepw[i >> 5] >> (i & 31)) & 1u) != 0u;
  float4 b = boxs[i];
  out[i * 5 + 0] = kept ? b.x : 0.0f;
  out[i * 5 + 1] = kept ? b.y : 0.0f;
  out[i * 5 + 2] = kept ? b.z : 0.0f;
  out[i * 5 + 3] = kept ? b.w : 0.0f;
  out[i * 5 + 4] = kept ? scrs[i] : 0.0f;
}

// ---------------------------------------------------------------- host launcher
extern "C" void kernel_launch(void* const* d_in, const int* in_sizes, int n_in,
                              void* d_out, int out_size, void* d_ws, size_t ws_size,
                              hipStream_t stream) {
  (void)in_sizes; (void)n_in; (void)out_size;
  const float* in = (const float*)d_in[0];
  float* out = (float*)d_out;
  char* ws = (char*)d_ws;

  unsigned long long* key = (unsigned long long*)(ws + OFF_KEY);
  float4*   xyxy = (float4*)  (ws + OFF_XYXY);
  float*    scr  = (float*)   (ws + OFF_SCORE);
  unsigned* vld  = (unsigned*)(ws + OFF_VALID);
  float4*   boxs = (float4*)  (ws + OFF_BOXS);
  float*    scrs = (float*)   (ws + OFF_SCRS);
  unsigned* vals = (unsigned*)(ws + OFF_VALS);
  unsigned* kpw  = (unsigned*)(ws + OFF_KEEPW);
  unsigned* mask = (unsigned*)(ws + OFF_MASK);

  // 1) transpose + keys
  k_prep<<<N_PAD / 256, 256, 0, stream>>>(in, key, xyxy, scr, vld);

  // 2) bitonic sort (ascending key == descending stable score order)
  //    global passes for j > 256, fused LDS kernel for the j <= 256 tail
  for (unsigned k = 2; k <= N_PAD; k <<= 1) {
    unsigned j = k >> 1;
    for (; j > 256; j >>= 1)
      k_bitonic<<<N_PAD / 256, 256, 0, stream>>>(key, j, k);
    k_bitonic_fused<<<N_PAD / 512, 256, 0, stream>>>(key, j, k);
  }

  // 3) gather into sorted order
  k_gather<<<(N_ANCHORS + 255) / 256, 256, 0, stream>>>(key, xyxy, scr, vld, boxs, scrs, vals);

  // 4+5) suppression: GPU-wide bitmask + single-wave serial scan (if ws fits), else direct
  if (ws_size >= WS_FULL) {
    dim3 g((N_ANCHORS + BI - 1) / BI, (N_ANCHORS + BJ - 1) / BJ);
    k_mask<<<g, 256, 0, stream>>>(boxs, mask);
    k_scan<<<1, 32, 0, stream>>>(mask, vals, kpw);
  } else {
    k_nms_direct<<<1, 1024, 0, stream>>>(boxs, vals, kpw);
  }

  // 6) emit (8400,5)
  k_write<<<(N_ANCHORS + 255) / 256, 256, 0, stream>>>(boxs, scrs, kpw, out);
}